// InteractionBlock_19207093748332
// MI455X (gfx1250) — compile-verified
//
#include <hip/hip_runtime.h>

// ---------------------------------------------------------------------------
// MACE-style interaction block on MI455X (gfx1250), fp32 WMMA everywhere.
// Weights are pre-packed into WMMA B-fragment order (one v2f load per frag).
// ---------------------------------------------------------------------------

typedef float v2f __attribute__((ext_vector_type(2)));
typedef float v8f __attribute__((ext_vector_type(8)));

#define NNODES 32768
#define NEDGES 524288

// D = A(16x4) * B(4x16) + C, fp32 matrix core op.
__device__ __forceinline__ v8f wmma4(v2f a, v2f b, v8f c) {
  return __builtin_amdgcn_wmma_f32_16x16x4_f32(
      /*neg_a=*/false, a, /*neg_b=*/false, b,
      /*c_mod=*/(short)0, c, /*reuse_a=*/false, /*reuse_b=*/false);
}

__device__ __forceinline__ float silu(float x) {
  return x / (1.0f + __expf(-x));
}

// ---------------------------------------------------------------------------
// Kernel: zero the scatter accumulators.
// ---------------------------------------------------------------------------
__global__ void k_zero(float* __restrict__ p, int n) {
  int i = blockIdx.x * blockDim.x + threadIdx.x;
  if (i < n) p[i] = 0.0f;
}

// ---------------------------------------------------------------------------
// Pack a row-major weight (nks*4 rows x nct*16 cols, leading dim ldb) into
// B-fragment order: P[(ks*nct+ct)*32 + lane] = {W[kk][col], W[kk+1][col]}
// with kk = ks*4 + 2*(lane>>4), col = ct*16 + (lane&15).
// ---------------------------------------------------------------------------
__global__ void k_pack_rm(const float* __restrict__ W, float* __restrict__ P,
                          int nks, int nct, int ldb) {
  const int idx = blockIdx.x * blockDim.x + threadIdx.x;
  const int n = nks * nct * 32;
  if (idx >= n) return;
  const int lane = idx & 31, fr = idx >> 5;
  const int ct = fr % nct, ks = fr / nct;
  const int kk = ks * 4 + 2 * (lane >> 4);
  const int col = ct * 16 + (lane & 15);
  v2f b;
  b.x = W[kk * ldb + col];
  b.y = W[(kk + 1) * ldb + col];
  ((v2f*)P)[idx] = b;
}

// ---------------------------------------------------------------------------
// Pack W_skip (64,64,10) into B' fragment order where conceptual
// B'[v][w*64+u] = Wsk[u*640 + v*10 + w]; tiles t = w*4+ug, nct = 40, nks = 16.
// ---------------------------------------------------------------------------
__global__ void k_pack_skip(const float* __restrict__ Wsk, float* __restrict__ P) {
  const int idx = blockIdx.x * blockDim.x + threadIdx.x;
  const int n = 16 * 40 * 32;
  if (idx >= n) return;
  const int lane = idx & 31, fr = idx >> 5;
  const int t = fr % 40, ks = fr / 40;
  const int w = t >> 2, ug = t & 3;
  const int u = ug * 16 + (lane & 15);
  const int kk = ks * 4 + 2 * (lane >> 4);
  v2f b;
  b.x = Wsk[u * 640 + kk * 10 + w];
  b.y = Wsk[u * 640 + (kk + 1) * 10 + w];
  ((v2f*)P)[idx] = b;
}

// Load 16 A fragments (K=64) from a row-major 16x64 LDS tile into registers.
__device__ __forceinline__ void ldsA16(const float* in, int lane, v2f* areg) {
  const int half = lane >> 4, r = lane & 15;
#pragma unroll
  for (int ks = 0; ks < 16; ++ks) {
    const int kk = ks * 4 + 2 * half;
    areg[ks].x = in[r * 64 + kk];
    areg[ks].y = in[r * 64 + kk + 1];
  }
}

// out(16x64) = act(A(16x64) @ B * sc) with A in registers, B packed.
// Fragment index = (ks*nct + ctBase + ct).
__device__ __forceinline__ void gemm64(const v2f* areg, const v2f* __restrict__ Pw,
                                       int nct, int ctBase, int lane, float sc,
                                       bool act, float* out) {
  const int half = lane >> 4, c = lane & 15;
  v8f acc[4] = {};
#pragma unroll
  for (int ks = 0; ks < 16; ++ks)
#pragma unroll
    for (int ct = 0; ct < 4; ++ct)
      acc[ct] = wmma4(areg[ks], Pw[(ks * nct + ctBase + ct) * 32 + lane], acc[ct]);
#pragma unroll
  for (int ct = 0; ct < 4; ++ct)
#pragma unroll
    for (int i = 0; i < 8; ++i) {
      const float v = acc[ct][i] * sc;
      out[(i + 8 * half) * 64 + ct * 16 + c] = act ? silu(v) : v;
    }
}

// ---------------------------------------------------------------------------
// Kernel: node up-projection. One wave per 16-node tile.
// ---------------------------------------------------------------------------
__global__ __launch_bounds__(256) void k_up(const float* __restrict__ nf,
                                            const v2f* __restrict__ Pu0,
                                            const v2f* __restrict__ Pu1,
                                            float* __restrict__ x0,
                                            float* __restrict__ x1) {
  const int lane = threadIdx.x & 31;
  const int wid  = (blockIdx.x * blockDim.x + threadIdx.x) >> 5;
  const int n0   = wid * 16;
  const int half = lane >> 4, r = lane & 15, c = lane & 15;
  const float sc = 0.125f;  // 1/sqrt(64)

  // ---- x0 = nf[:, 0:64] @ Wu0 ----
  {
    v8f acc[4] = {};
    for (int ks = 0; ks < 16; ++ks) {
      const int kk = ks * 4 + 2 * half;
      v2f a;
      a.x = nf[(n0 + r) * 256 + kk];
      a.y = nf[(n0 + r) * 256 + kk + 1];
#pragma unroll
      for (int ct = 0; ct < 4; ++ct)
        acc[ct] = wmma4(a, Pu0[(ks * 4 + ct) * 32 + lane], acc[ct]);
    }
#pragma unroll
    for (int ct = 0; ct < 4; ++ct)
#pragma unroll
      for (int i = 0; i < 8; ++i)
        x0[(n0 + i + 8 * half) * 64 + ct * 16 + c] = acc[ct][i] * sc;
  }

  // ---- x1 components, A read with element stride 3 ----
  for (int d3 = 0; d3 < 3; ++d3) {
    v8f acc[4] = {};
    for (int ks = 0; ks < 16; ++ks) {
      const int kk = ks * 4 + 2 * half;
      v2f a;
      a.x = nf[(n0 + r) * 256 + 64 + kk * 3 + d3];
      a.y = nf[(n0 + r) * 256 + 64 + (kk + 1) * 3 + d3];
#pragma unroll
      for (int ct = 0; ct < 4; ++ct)
        acc[ct] = wmma4(a, Pu1[(ks * 4 + ct) * 32 + lane], acc[ct]);
    }
#pragma unroll
    for (int ct = 0; ct < 4; ++ct)
#pragma unroll
      for (int i = 0; i < 8; ++i)
        x1[(n0 + i + 8 * half) * 192 + (ct * 16 + c) * 3 + d3] = acc[ct][i] * sc;
  }
}

// ---------------------------------------------------------------------------
// Kernel: fused edge MLP + message formation + atomic scatter.
// One wave per 16-edge tile, 2 waves per block (LDS: ~20.5KB per wave).
// All four tpw groups kept in LDS -> single scatter pass per tile.
// ---------------------------------------------------------------------------
__global__ __launch_bounds__(64) void k_edge(
    const float* __restrict__ ef,   // (E,8)
    const float* __restrict__ ea,   // (E,4)
    const int* __restrict__ src, const int* __restrict__ dst,
    const v2f* __restrict__ Pr1,    // packed (8,64)
    const v2f* __restrict__ Pr2,    // packed (64,64)
    const v2f* __restrict__ Pr3,    // packed (64,64)
    const v2f* __restrict__ Pr4,    // packed (64,256), nct=16
    const float* __restrict__ x0,   // (N,64)
    const float* __restrict__ x1,   // (N,192)
    float* __restrict__ m0,         // (N,128)
    float* __restrict__ m1) {       // (N,384)
  __shared__ float hst_s[2][1024];   // layer staging (h1/h3)
  __shared__ float tbuf_s[2][4096];  // h2 (slot 0, transient) + tpw groups 0..3
  __shared__ float sea_s[2][64];     // edge_attrs tile
  __shared__ int   ssrc_s[2][16];
  __shared__ int   sdst_s[2][16];

  const int lane = threadIdx.x & 31;
  const int lwid = threadIdx.x >> 5;
  float* hst  = hst_s[lwid];
  float* tbuf = tbuf_s[lwid];
  float* sea  = sea_s[lwid];
  int*   ssrc = ssrc_s[lwid];
  int*   sdst = sdst_s[lwid];

  const int wid  = (blockIdx.x * blockDim.x + threadIdx.x) >> 5;
  const int e0   = wid * 16;
  const int half = lane >> 4, r = lane & 15, c = lane & 15;

  // ---- cache per-edge metadata in LDS ----
  if (lane < 16) ssrc[lane] = src[e0 + lane];
  else           sdst[lane - 16] = dst[e0 + lane - 16];
  sea[lane]      = ea[e0 * 4 + lane];
  sea[lane + 32] = ea[e0 * 4 + lane + 32];

  // ---- layer 1: ef(16x8) @ Wr1 * 1/sqrt(8), silu -> hst ----
  {
    const float sc = 0.35355339059327373f;
    v8f acc[4] = {};
    for (int ks = 0; ks < 2; ++ks) {
      const int kk = ks * 4 + 2 * half;
      v2f a;
      a.x = ef[(e0 + r) * 8 + kk];
      a.y = ef[(e0 + r) * 8 + kk + 1];
#pragma unroll
      for (int ct = 0; ct < 4; ++ct)
        acc[ct] = wmma4(a, Pr1[(ks * 4 + ct) * 32 + lane], acc[ct]);
    }
#pragma unroll
    for (int ct = 0; ct < 4; ++ct)
#pragma unroll
      for (int i = 0; i < 8; ++i)
        hst[(i + 8 * half) * 64 + ct * 16 + c] = silu(acc[ct][i] * sc);
  }

  v2f areg[16];
  // ---- layer 2: h1(hst) -> h2(tbuf[0:1024], transient) ----
  ldsA16(hst, lane, areg);
  gemm64(areg, Pr2, 4, 0, lane, 0.125f, true, tbuf);
  // ---- layer 3: h2 -> h3(hst) ----
  ldsA16(tbuf, lane, areg);
  gemm64(areg, Pr3, 4, 0, lane, 0.125f, true, hst);

  // ---- layer 4: all four 64-col groups -> tbuf[g*1024 ...] ----
  ldsA16(hst, lane, areg);
  for (int g = 0; g < 4; ++g)
    gemm64(areg, Pr4, 16, g * 4, lane, 0.125f, false, tbuf + g * 1024);

  // ---- single scatter pass: 16 edges x 64 channels over 32 lanes ----
  const float sq3inv = 0.57735026918962584f;
  for (int it = 0; it < 32; ++it) {
    const int p  = it * 32 + lane;
    const int el = p >> 6;
    const int u  = p & 63;
    const float tp0 = tbuf[el * 64 + u];
    const float tp1 = tbuf[1024 + el * 64 + u];
    const float tp2 = tbuf[2048 + el * 64 + u];
    const float tp3 = tbuf[3072 + el * 64 + u];
    const int sn = ssrc[el];
    const int dn = sdst[el];
    const float sh0  = sea[el * 4 + 0];
    const float sh1x = sea[el * 4 + 1];
    const float sh1y = sea[el * 4 + 2];
    const float sh1z = sea[el * 4 + 3];
    const float s  = x0[sn * 64 + u];
    const float vx = x1[sn * 192 + u * 3 + 0];
    const float vy = x1[sn * 192 + u * 3 + 1];
    const float vz = x1[sn * 192 + u * 3 + 2];

    unsafeAtomicAdd(&m0[dn * 128 + u], tp0 * s * sh0);
    unsafeAtomicAdd(&m0[dn * 128 + 64 + u],
                    tp3 * (vx * sh1x + vy * sh1y + vz * sh1z) * sq3inv);
    const float p1s = tp1 * s;
    unsafeAtomicAdd(&m1[dn * 384 + u * 3 + 0], p1s * sh1x);
    unsafeAtomicAdd(&m1[dn * 384 + u * 3 + 1], p1s * sh1y);
    unsafeAtomicAdd(&m1[dn * 384 + u * 3 + 2], p1s * sh1z);
    const float p2s = tp2 * sh0;
    unsafeAtomicAdd(&m1[dn * 384 + 192 + u * 3 + 0], p2s * vx);
    unsafeAtomicAdd(&m1[dn * 384 + 192 + u * 3 + 1], p2s * vy);
    unsafeAtomicAdd(&m1[dn * 384 + 192 + u * 3 + 2], p2s * vz);
  }
}

// ---------------------------------------------------------------------------
// Kernel: per-node linear + element-dependent skip contraction.
//   y = m @ Wl * scale ; z[n,u] = inv * sum_w attrs[n,w] * G[n, w*64+u]
// ---------------------------------------------------------------------------
__global__ __launch_bounds__(256) void k_node(
    const float* __restrict__ m0,   // (N,128)
    const float* __restrict__ m1,   // (N,384)
    const float* __restrict__ na,   // (N,10)
    const v2f* __restrict__ Pl0,    // packed (128,64), nct=4, nks=32
    const v2f* __restrict__ Pl1,
    const v2f* __restrict__ Psk0,   // packed skip, nct=40, nks=16
    const v2f* __restrict__ Psk1,
    float* __restrict__ out) {      // (N,256)
  __shared__ float lds[8][1024 + 160];
  const int lane = threadIdx.x & 31;
  const int lwid = threadIdx.x >> 5;
  float* ybuf = &lds[lwid][0];
  float* abuf = &lds[lwid][1024];
  const int wid  = (blockIdx.x * blockDim.x + threadIdx.x) >> 5;
  const int n0   = wid * 16;
  const int half = lane >> 4, r = lane & 15, c = lane & 15;
  const float scale = 0.0055242717280199028f;  // 1/(sqrt(128)*16)
  const float inv   = 0.039528470752104741f;   // 1/sqrt(640)

  for (int idx = lane; idx < 160; idx += 32)
    abuf[idx] = na[(n0 + idx / 10) * 10 + idx % 10];

  for (int phase = 0; phase < 4; ++phase) {
    // ---- y -> ybuf ----
    {
      const v2f* Pl = (phase == 0) ? Pl0 : Pl1;
      v8f acc[4] = {};
      for (int ks = 0; ks < 32; ++ks) {
        const int kk = ks * 4 + 2 * half;
        v2f a;
        if (phase == 0) {
          a.x = m0[(n0 + r) * 128 + kk];
          a.y = m0[(n0 + r) * 128 + kk + 1];
        } else {
          const int i2 = phase - 1;
          a.x = m1[(n0 + r) * 384 + kk * 3 + i2];
          a.y = m1[(n0 + r) * 384 + (kk + 1) * 3 + i2];
        }
#pragma unroll
        for (int ct = 0; ct < 4; ++ct)
          acc[ct] = wmma4(a, Pl[(ks * 4 + ct) * 32 + lane], acc[ct]);
      }
#pragma unroll
      for (int ct = 0; ct < 4; ++ct)
#pragma unroll
        for (int i = 0; i < 8; ++i)
          ybuf[(i + 8 * half) * 64 + ct * 16 + c] = acc[ct][i] * scale;
    }

    // ---- skip contraction: A fragments of y held in registers ----
    const v2f* Psk = (phase == 0) ? Psk0 : Psk1;
    v2f areg[16];
    ldsA16(ybuf, lane, areg);

    float zacc[4][8];
#pragma unroll
    for (int ug = 0; ug < 4; ++ug)
#pragma unroll
      for (int i = 0; i < 8; ++i) zacc[ug][i] = 0.0f;

    for (int w = 0; w < 10; ++w) {
      v8f acc[4] = {};
#pragma unroll
      for (int ks = 0; ks < 16; ++ks)
#pragma unroll
        for (int ug = 0; ug < 4; ++ug)
          acc[ug] = wmma4(areg[ks], Psk[(ks * 40 + w * 4 + ug) * 32 + lane], acc[ug]);
#pragma unroll
      for (int ug = 0; ug < 4; ++ug)
#pragma unroll
        for (int i = 0; i < 8; ++i)
          zacc[ug][i] += abuf[(i + 8 * half) * 10 + w] * acc[ug][i];
    }

    // ---- write z ----
#pragma unroll
    for (int ug = 0; ug < 4; ++ug)
#pragma unroll
      for (int i = 0; i < 8; ++i) {
        const int n = n0 + i + 8 * half;
        const int u = ug * 16 + c;
        const float z = zacc[ug][i] * inv;
        if (phase == 0) out[n * 256 + u] = z;
        else            out[n * 256 + 64 + u * 3 + (phase - 1)] = z;
      }
  }
}

// ---------------------------------------------------------------------------
extern "C" void kernel_launch(void* const* d_in, const int* in_sizes, int n_in,
                              void* d_out, int out_size, void* d_ws, size_t ws_size,
                              hipStream_t stream) {
  const float* nf   = (const float*)d_in[0];
  const float* na   = (const float*)d_in[1];
  const float* ef   = (const float*)d_in[2];
  const float* ea   = (const float*)d_in[3];
  const int*   src  = (const int*)d_in[4];
  const int*   dst  = (const int*)d_in[5];
  const float* Wu0  = (const float*)d_in[6];
  const float* Wu1  = (const float*)d_in[7];
  const float* Wr1  = (const float*)d_in[8];
  const float* Wr2  = (const float*)d_in[9];
  const float* Wr3  = (const float*)d_in[10];
  const float* Wr4  = (const float*)d_in[11];
  const float* Wl0  = (const float*)d_in[12];
  const float* Wl1  = (const float*)d_in[13];
  const float* Wsk0 = (const float*)d_in[14];
  const float* Wsk1 = (const float*)d_in[15];
  float* out = (float*)d_out;

  float* ws = (float*)d_ws;
  float* x0 = ws;                              // N*64
  float* x1 = ws + (size_t)NNODES * 64;        // N*192
  float* m0 = ws + (size_t)NNODES * 256;       // N*128
  float* m1 = ws + (size_t)NNODES * 384;       // N*384

  // Packed weights after the m-buffers.
  float* pp   = ws + (size_t)NNODES * 768;
  float* PWu0 = pp;            pp += 16 * 4 * 64;    // 4096
  float* PWu1 = pp;            pp += 16 * 4 * 64;
  float* PWr1 = pp;            pp += 2 * 4 * 64;     // 512
  float* PWr2 = pp;            pp += 16 * 4 * 64;
  float* PWr3 = pp;            pp += 16 * 4 * 64;
  float* PWr4 = pp;            pp += 16 * 16 * 64;   // 16384
  float* PWl0 = pp;            pp += 32 * 4 * 64;    // 8192
  float* PWl1 = pp;            pp += 32 * 4 * 64;
  float* Psk0 = pp;            pp += 16 * 40 * 64;   // 40960
  float* Psk1 = pp;            pp += 16 * 40 * 64;

  // ---- weight packing (tiny) ----
  k_pack_rm<<<(16 * 4 * 32 + 255) / 256, 256, 0, stream>>>(Wu0, PWu0, 16, 4, 64);
  k_pack_rm<<<(16 * 4 * 32 + 255) / 256, 256, 0, stream>>>(Wu1, PWu1, 16, 4, 64);
  k_pack_rm<<<(2 * 4 * 32 + 255) / 256, 256, 0, stream>>>(Wr1, PWr1, 2, 4, 64);
  k_pack_rm<<<(16 * 4 * 32 + 255) / 256, 256, 0, stream>>>(Wr2, PWr2, 16, 4, 64);
  k_pack_rm<<<(16 * 4 * 32 + 255) / 256, 256, 0, stream>>>(Wr3, PWr3, 16, 4, 64);
  k_pack_rm<<<(16 * 16 * 32 + 255) / 256, 256, 0, stream>>>(Wr4, PWr4, 16, 16, 256);
  k_pack_rm<<<(32 * 4 * 32 + 255) / 256, 256, 0, stream>>>(Wl0, PWl0, 32, 4, 64);
  k_pack_rm<<<(32 * 4 * 32 + 255) / 256, 256, 0, stream>>>(Wl1, PWl1, 32, 4, 64);
  k_pack_skip<<<(16 * 40 * 32 + 255) / 256, 256, 0, stream>>>(Wsk0, Psk0);
  k_pack_skip<<<(16 * 40 * 32 + 255) / 256, 256, 0, stream>>>(Wsk1, Psk1);

  // ---- compute pipeline ----
  const int nz = NNODES * 512;  // m0 + m1
  k_zero<<<(nz + 255) / 256, 256, 0, stream>>>(m0, nz);
  k_up<<<NNODES / 16 / 8, 256, 0, stream>>>(nf, (const v2f*)PWu0, (const v2f*)PWu1, x0, x1);
  k_edge<<<NEDGES / 16 / 2, 64, 0, stream>>>(ef, ea, src, dst,
                                             (const v2f*)PWr1, (const v2f*)PWr2,
                                             (const v2f*)PWr3, (const v2f*)PWr4,
                                             x0, x1, m0, m1);
  k_node<<<NNODES / 16 / 8, 256, 0, stream>>>(m0, m1, na,
                                              (const v2f*)PWl0, (const v2f*)PWl1,
                                              (const v2f*)Psk0, (const v2f*)Psk1,
                                              out);
}